// Graph_78400333021487
// MI455X (gfx1250) — compile-verified
//
#include <hip/hip_runtime.h>
#include <math.h>

#define DEV_INLINE __device__ __forceinline__

namespace {
constexpr int B = 4, L = 192, E = 192, H = 128;
constexpr int NH = 8, HD = 16, P = 128;
constexpr int ITERS = 3, BMES = 10, LEN = 10;
constexpr float INV_SCALE = 0.25f;      // 1/sqrt(HD)
constexpr int RB = B * 192;             // 768 rows (B*L == B*E)
constexpr long RH = (long)RB * H;       // 98304

typedef __attribute__((ext_vector_type(2))) float v2f;
typedef __attribute__((ext_vector_type(8))) float v8f;

DEV_INLINE float sigm(float x) { return 1.0f / (1.0f + __expf(-x)); }
DEV_INLINE float waveMax(float v) {
  for (int o = 16; o > 0; o >>= 1) v = fmaxf(v, __shfl_xor(v, o, 32));
  return v;
}
DEV_INLINE float waveSum(float v) {
  for (int o = 16; o > 0; o >>= 1) v += __shfl_xor(v, o, 32);
  return v;
}

// ---------------------------------------------------------------------------
// Tiled GEMM: C[M x 128] = A[M x K] @ W[K x 128] (+ bias) using fp32 WMMA.
// One wave32 computes one 16x16 tile with v_wmma_f32_16x16x4_f32.
// A layout (16x4 fp32): lane%16 = M row, lane/16 selects K pair {0,1}/{2,3}.
// B layout (4x16 fp32): lane%16 = N col, lane/16 selects K pair.
// grid.z batches gate matrices via strides.
// ---------------------------------------------------------------------------
template <bool ACC>
__global__ __launch_bounds__(32) void gemm16(
    const float* __restrict__ A, const float* __restrict__ W,
    const float* __restrict__ bias, float* __restrict__ C, int K,
    long wStride, long bStride, long cStride) {
  const int n0 = blockIdx.x * 16;
  const int m0 = blockIdx.y * 16;
  const int g = blockIdx.z;
  W += (long)g * wStride;
  C += (long)g * cStride;
  const int lane = threadIdx.x;
  const int lm = lane & 15;
  const int half = lane >> 4;
  const float* arow = A + (long)(m0 + lm) * K;
  v8f acc = {};
  for (int k = 0; k < K; k += 4) {
    const int ka = k + 2 * half;
    v2f av, bv;
    av.x = arow[ka];
    av.y = arow[ka + 1];
    bv.x = W[(long)ka * P + n0 + lm];
    bv.y = W[(long)(ka + 1) * P + n0 + lm];
    acc = __builtin_amdgcn_wmma_f32_16x16x4_f32(false, av, false, bv, (short)0,
                                                acc, false, false);
  }
  float bn = 0.0f;
  if (bias) bn = (bias + (long)g * bStride)[n0 + lm];
#pragma unroll
  for (int j = 0; j < 8; ++j) {
    const int row = m0 + half * 8 + j;
    float v = acc[j] + bn;
    float* dst = &C[(long)row * P + n0 + lm];
    if (ACC) *dst += v; else *dst = v;
  }
}

// ---------------------------------------------------------------------------
// LayerNorm over H=128 (one block per row).
// ---------------------------------------------------------------------------
__global__ __launch_bounds__(128) void ln_rows(const float* __restrict__ x,
                                               const float* __restrict__ g,
                                               const float* __restrict__ b,
                                               float* __restrict__ out) {
  __shared__ float red[128];
  const int row = blockIdx.x, t = threadIdx.x;
  const float v = x[(long)row * H + t];
  red[t] = v; __syncthreads();
  for (int s = 64; s > 0; s >>= 1) { if (t < s) red[t] += red[t + s]; __syncthreads(); }
  const float m = red[0] * (1.0f / H); __syncthreads();
  const float d = v - m;
  red[t] = d * d; __syncthreads();
  for (int s = 64; s > 0; s >>= 1) { if (t < s) red[t] += red[t + s]; __syncthreads(); }
  const float var = red[0] * (1.0f / H);
  out[(long)row * H + t] = d * rsqrtf(var + 1e-5f) * g[t] + b[t];
}

__global__ __launch_bounds__(128) void glo_mean(const float* __restrict__ nodes,
                                                const float* __restrict__ edges,
                                                float* __restrict__ glo) {
  const int b = blockIdx.x, h = threadIdx.x;
  float s = 0.0f;
  for (int i = 0; i < L; ++i) s += nodes[((long)b * L + i) * H + h];
  for (int i = 0; i < E; ++i) s += edges[((long)b * E + i) * H + h];
  glo[b * H + h] = s * (1.0f / (L + E));
}

// ---------------------------------------------------------------------------
// Fused n2e MHA per (b,e): K/V = precomputed node projections + on-the-fly
// BMES (K=10) correction held in 212KB of LDS (CDNA5: 320KB/WGP), then
// masked softmax over L, AV, WO projection, leaky-relu, residual, LayerNorm.
// ---------------------------------------------------------------------------
__global__ __launch_bounds__(256) void attn_n2e(
    const float* __restrict__ q, const float* __restrict__ kn,
    const float* __restrict__ vn, const float* __restrict__ bmes,
    const float* __restrict__ WK2, const float* __restrict__ WV2,
    const int* __restrict__ mask,  // (B,L,E)
    const float* __restrict__ WO, const float* __restrict__ bO,
    const float* __restrict__ lng, const float* __restrict__ lnb,
    const float* __restrict__ resid, float* __restrict__ out) {
  extern __shared__ float smem[];
  float* s_k = smem;                 // L*P
  float* s_v = s_k + L * P;          // L*P
  float* s_q = s_v + L * P;          // P
  float* s_al = s_q + P;             // NH*L
  float* s_att = s_al + NH * L;      // P
  float* s_red = s_att + P;          // 128
  float* s_bm = s_red + 128;         // L*BMES
  const int be = blockIdx.x, b = be / E, e = be % E;
  const int tid = threadIdx.x;
  for (int i = tid; i < L * BMES; i += 256)
    s_bm[i] = bmes[(long)be * L * BMES + i];
  for (int i = tid; i < P; i += 256) s_q[i] = q[(long)be * P + i];
  __syncthreads();
  for (int i = tid; i < L * P; i += 256) {
    const int l = i >> 7, p = i & 127;
    float ak = kn[((long)b * L + l) * P + p];
    float av = vn[((long)b * L + l) * P + p];
#pragma unroll
    for (int j = 0; j < BMES; ++j) {
      const float t = s_bm[l * BMES + j];
      ak += t * WK2[j * P + p];
      av += t * WV2[j * P + p];
    }
    s_k[i] = ak; s_v[i] = av;
  }
  __syncthreads();
  const int n = tid >> 5, lane = tid & 31;
  float al[L / 32];
  float mx = -3.0e38f;
#pragma unroll
  for (int c = 0; c < L / 32; ++c) {
    const int l = lane + 32 * c;
    float s = 0.0f;
#pragma unroll
    for (int d = 0; d < HD; ++d) s += s_q[n * HD + d] * s_k[l * P + n * HD + d];
    s *= INV_SCALE;
    if (mask[((long)b * L + l) * E + e]) s = -3.0e38f;
    al[c] = s; mx = fmaxf(mx, s);
  }
  mx = waveMax(mx);
  float sum = 0.0f;
#pragma unroll
  for (int c = 0; c < L / 32; ++c) {
    const float ev = (al[c] <= -1.0e38f) ? 0.0f : __expf(al[c] - mx);
    al[c] = ev; sum += ev;
  }
  sum = waveSum(sum);
  const float inv = sum > 0.0f ? 1.0f / sum : 0.0f;
#pragma unroll
  for (int c = 0; c < L / 32; ++c) s_al[n * L + lane + 32 * c] = al[c] * inv;
  __syncthreads();
  if (tid < P) {
    const int nh = tid >> 4;
    float a = 0.0f;
    for (int l = 0; l < L; ++l) a += s_al[nh * L + l] * s_v[l * P + tid];
    s_att[tid] = a;
  }
  __syncthreads();
  float o = 0.0f;
  if (tid < H) {
    o = bO[tid];
    for (int p = 0; p < P; ++p) o += s_att[p] * WO[p * H + tid];
    o = o > 0.0f ? o : 0.01f * o;
    o += resid[(long)be * H + tid];
    s_red[tid] = o;
  }
  __syncthreads();
  for (int s = 64; s > 0; s >>= 1) { if (tid < s) s_red[tid] += s_red[tid + s]; __syncthreads(); }
  const float m = s_red[0] * (1.0f / H); __syncthreads();
  if (tid < H) { const float d = o - m; s_red[tid] = d * d; }
  __syncthreads();
  for (int s = 64; s > 0; s >>= 1) { if (tid < s) s_red[tid] += s_red[tid + s]; __syncthreads(); }
  const float var = s_red[0] * (1.0f / H);
  if (tid < H)
    out[(long)be * H + tid] = (o - m) * rsqrtf(var + 1e-5f) * lng[tid] + lnb[tid];
}

// ---------------------------------------------------------------------------
// e2n MHA per (b,l): values are l-independent, so this is plain masked
// attention over E with precomputed K/V in L2.
// ---------------------------------------------------------------------------
__global__ __launch_bounds__(256) void attn_e2n(
    const float* __restrict__ q, const float* __restrict__ ke,
    const float* __restrict__ ve,
    const int* __restrict__ mask,  // (B,E,L)
    const float* __restrict__ WO, const float* __restrict__ bO,
    const float* __restrict__ lng, const float* __restrict__ lnb,
    const float* __restrict__ resid, float* __restrict__ out) {
  __shared__ float s_q[P];
  __shared__ float s_al[NH * E];
  __shared__ float s_att[P];
  __shared__ float s_red[128];
  const int bl = blockIdx.x, b = bl / L, l = bl % L;
  const int tid = threadIdx.x;
  for (int i = tid; i < P; i += 256) s_q[i] = q[(long)bl * P + i];
  __syncthreads();
  const int n = tid >> 5, lane = tid & 31;
  float al[E / 32];
  float mx = -3.0e38f;
#pragma unroll
  for (int c = 0; c < E / 32; ++c) {
    const int e = lane + 32 * c;
    float s = 0.0f;
#pragma unroll
    for (int d = 0; d < HD; ++d)
      s += s_q[n * HD + d] * ke[((long)b * E + e) * P + n * HD + d];
    s *= INV_SCALE;
    if (mask[((long)b * E + e) * L + l]) s = -3.0e38f;
    al[c] = s; mx = fmaxf(mx, s);
  }
  mx = waveMax(mx);
  float sum = 0.0f;
#pragma unroll
  for (int c = 0; c < E / 32; ++c) {
    const float ev = (al[c] <= -1.0e38f) ? 0.0f : __expf(al[c] - mx);
    al[c] = ev; sum += ev;
  }
  sum = waveSum(sum);
  const float inv = sum > 0.0f ? 1.0f / sum : 0.0f;
#pragma unroll
  for (int c = 0; c < E / 32; ++c) s_al[n * E + lane + 32 * c] = al[c] * inv;
  __syncthreads();
  if (tid < P) {
    const int nh = tid >> 4;
    float a = 0.0f;
    for (int e = 0; e < E; ++e) a += s_al[nh * E + e] * ve[((long)b * E + e) * P + tid];
    s_att[tid] = a;
  }
  __syncthreads();
  float o = 0.0f;
  if (tid < H) {
    o = bO[tid];
    for (int p = 0; p < P; ++p) o += s_att[p] * WO[p * H + tid];
    o = o > 0.0f ? o : 0.01f * o;
    o += resid[(long)bl * H + tid];
    s_red[tid] = o;
  }
  __syncthreads();
  for (int s = 64; s > 0; s >>= 1) { if (tid < s) s_red[tid] += s_red[tid + s]; __syncthreads(); }
  const float m = s_red[0] * (1.0f / H); __syncthreads();
  if (tid < H) { const float d = o - m; s_red[tid] = d * d; }
  __syncthreads();
  for (int s = 64; s > 0; s >>= 1) { if (tid < s) s_red[tid] += s_red[tid + s]; __syncthreads(); }
  const float var = s_red[0] * (1.0f / H);
  if (tid < H)
    out[(long)bl * H + tid] = (o - m) * rsqrtf(var + 1e-5f) * lng[tid] + lnb[tid];
}

// ---------------------------------------------------------------------------
// Global-state attention (1 query per batch over 192 rows).
// ---------------------------------------------------------------------------
__global__ __launch_bounds__(256) void glo_att(
    const float* __restrict__ glo, const float* __restrict__ kg,
    const float* __restrict__ vg, const float* __restrict__ WQ,
    const float* __restrict__ bQ, const float* __restrict__ WO,
    const float* __restrict__ bO, const float* __restrict__ lng,
    const float* __restrict__ lnb, float* __restrict__ gatt, int outOff) {
  constexpr int N = 192;
  __shared__ float s_q[P];
  __shared__ float s_al[NH * N];
  __shared__ float s_att[P];
  __shared__ float s_red[128];
  const int b = blockIdx.x, tid = threadIdx.x;
  if (tid < P) {
    float s = bQ[tid];
    for (int k = 0; k < H; ++k) s += glo[b * H + k] * WQ[k * P + tid];
    s_q[tid] = s;
  }
  __syncthreads();
  const int n = tid >> 5, lane = tid & 31;
  float al[N / 32];
  float mx = -3.0e38f;
#pragma unroll
  for (int c = 0; c < N / 32; ++c) {
    const int e = lane + 32 * c;
    float s = 0.0f;
#pragma unroll
    for (int d = 0; d < HD; ++d)
      s += s_q[n * HD + d] * kg[((long)b * N + e) * P + n * HD + d];
    al[c] = s * INV_SCALE; mx = fmaxf(mx, al[c]);
  }
  mx = waveMax(mx);
  float sum = 0.0f;
#pragma unroll
  for (int c = 0; c < N / 32; ++c) { al[c] = __expf(al[c] - mx); sum += al[c]; }
  sum = waveSum(sum);
  const float inv = 1.0f / sum;
#pragma unroll
  for (int c = 0; c < N / 32; ++c) s_al[n * N + lane + 32 * c] = al[c] * inv;
  __syncthreads();
  if (tid < P) {
    const int nh = tid >> 4;
    float a = 0.0f;
    for (int e = 0; e < N; ++e) a += s_al[nh * N + e] * vg[((long)b * N + e) * P + tid];
    s_att[tid] = a;
  }
  __syncthreads();
  float o = 0.0f;
  if (tid < H) {
    o = bO[tid];
    for (int p = 0; p < P; ++p) o += s_att[p] * WO[p * H + tid];
    o = o > 0.0f ? o : 0.01f * o;
    o += glo[b * H + tid];
    s_red[tid] = o;
  }
  __syncthreads();
  for (int s = 64; s > 0; s >>= 1) { if (tid < s) s_red[tid] += s_red[tid + s]; __syncthreads(); }
  const float m = s_red[0] * (1.0f / H); __syncthreads();
  if (tid < H) { const float d = o - m; s_red[tid] = d * d; }
  __syncthreads();
  for (int s = 64; s > 0; s >>= 1) { if (tid < s) s_red[tid] += s_red[tid + s]; __syncthreads(); }
  const float var = s_red[0] * (1.0f / H);
  if (tid < H)
    gatt[b * 2 * H + outOff + tid] =
        (o - m) * rsqrtf(var + 1e-5f) * lng[tid] + lnb[tid];
}

// words_length (K=10) contribution to e2n K/V.
__global__ __launch_bounds__(128) void lenproj_add(
    const float* __restrict__ wl, const float* __restrict__ WK3,
    const float* __restrict__ WV3, float* __restrict__ kbuf,
    float* __restrict__ vbuf) {
  const int row = blockIdx.x, p = threadIdx.x;
  float ak = 0.0f, av = 0.0f;
#pragma unroll
  for (int j = 0; j < LEN; ++j) {
    const float t = wl[(long)row * LEN + j];
    ak += t * WK3[j * P + p];
    av += t * WV3[j * P + p];
  }
  kbuf[(long)row * P + p] += ak;
  vbuf[(long)row * P + p] += av;
}

__global__ __launch_bounds__(256) void cat_edge(const float* __restrict__ edges,
                                                const float* __restrict__ eatt,
                                                const float* __restrict__ gatt,
                                                float* __restrict__ cat) {
  const long i = blockIdx.x * 256L + threadIdx.x;
  if (i >= (long)RB * 512) return;
  const long row = i >> 9; const int c = (int)(i & 511);
  const int b = (int)(row / E);
  float v;
  if (c < 128) v = edges[row * 128 + c];
  else if (c < 256) v = eatt[row * 128 + (c - 128)];
  else v = gatt[b * 256 + (c - 256)];
  cat[i] = v;
}

__global__ __launch_bounds__(256) void cat_node(const float* __restrict__ nodes,
                                                const float* __restrict__ ce,
                                                const float* __restrict__ natt,
                                                const float* __restrict__ gatt,
                                                float* __restrict__ cat) {
  const long i = blockIdx.x * 256L + threadIdx.x;
  if (i >= (long)RB * 640) return;
  const long row = i / 640; const int c = (int)(i % 640);
  const int b = (int)(row / L);
  float v;
  if (c < 128) v = nodes[row * 128 + c];
  else if (c < 256) v = ce[row * 128 + (c - 128)];
  else if (c < 384) v = natt[row * 128 + (c - 256)];
  else v = gatt[b * 256 + (c - 384)];
  cat[i] = v;
}

__global__ __launch_bounds__(128) void edge_combine(const float* __restrict__ gates,
                                                    float* __restrict__ edges,
                                                    float* __restrict__ edges_ri) {
  const long idx = blockIdx.x * 128L + threadIdx.x;
  const float si = sigm(gates[idx]);
  const float sf = sigm(gates[RH + idx]);
  const float c = tanhf(gates[2 * RH + idx]);
  const float m = fmaxf(si, sf);
  const float ei = __expf(si - m), ef = __expf(sf - m);
  const float h = edges[idx];
  const float nv = (ei * c + ef * h) / (ei + ef);
  edges[idx] = nv; edges_ri[idx] = nv;
}

__global__ __launch_bounds__(128) void node_combine(const float* __restrict__ gates,
                                                    const float* __restrict__ natt,
                                                    float* __restrict__ nodes,
                                                    float* __restrict__ nodes_ri) {
  const long idx = blockIdx.x * 128L + threadIdx.x;
  const float a0 = sigm(gates[idx]);
  const float a1 = sigm(gates[RH + idx]);
  const float a2 = sigm(gates[2 * RH + idx]);
  const float cx = tanhf(gates[3 * RH + idx]);
  const float m = fmaxf(a0, fmaxf(a1, a2));
  const float e0 = __expf(a0 - m), e1 = __expf(a1 - m), e2 = __expf(a2 - m);
  const float inv = 1.0f / (e0 + e1 + e2);
  const float nv = (e0 * cx + e1 * natt[idx] + e2 * nodes[idx]) * inv;
  nodes[idx] = nv; nodes_ri[idx] = nv;
}

__global__ __launch_bounds__(128) void glo_cell(const float* __restrict__ gatt,
                                                const float* __restrict__ W,
                                                const float* __restrict__ bias,
                                                float* __restrict__ glo) {
  __shared__ float s_g[H];
  __shared__ float s_a[2 * H];
  const int b = blockIdx.x, h = threadIdx.x;
  s_g[h] = glo[b * H + h];
  s_a[h] = gatt[b * 2 * H + h];
  s_a[H + h] = gatt[b * 2 * H + H + h];
  __syncthreads();
  float g3[3];
#pragma unroll
  for (int gi = 0; gi < 3; ++gi) {
    const float* Wg = W + (long)gi * 384 * H;
    float s = bias[gi * H + h];
    for (int k = 0; k < H; ++k) s += s_g[k] * Wg[k * H + h];
    for (int k = 0; k < 2 * H; ++k) s += s_a[k] * Wg[(H + k) * H + h];
    g3[gi] = s;
  }
  const float i = sigm(g3[0]), f = sigm(g3[1]), c = tanhf(g3[2]);
  const float m = fmaxf(i, f);
  const float ei = __expf(i - m), ef = __expf(f - m);
  glo[b * H + h] = (ei * c + ef * s_g[h]) / (ei + ef);
}

__global__ __launch_bounds__(128) void layer_att(const float* __restrict__ stack,
                                                 const float* __restrict__ Wv,
                                                 const float* __restrict__ bv,
                                                 float* __restrict__ out, int R) {
  __shared__ float s_red[128];
  __shared__ float s_dot[ITERS];
  const int br = blockIdx.x, b = br / R, r = br % R, t = threadIdx.x;
  const long istride = (long)B * R * H;
  const long base = ((long)b * R + r) * H + t;
  const float x0 = stack[base], x1 = stack[base + istride],
              x2 = stack[base + 2 * istride];
  const float w = Wv[t];
#pragma unroll
  for (int i = 0; i < ITERS; ++i) {
    const float xi = (i == 0) ? x0 : ((i == 1) ? x1 : x2);
    s_red[t] = xi * w; __syncthreads();
    for (int s = 64; s > 0; s >>= 1) { if (t < s) s_red[t] += s_red[t + s]; __syncthreads(); }
    if (t == 0) s_dot[i] = s_red[0] + bv[0];
    __syncthreads();
  }
  const float d0 = s_dot[0], d1 = s_dot[1], d2 = s_dot[2];
  const float m = fmaxf(d0, fmaxf(d1, d2));
  const float e0 = __expf(d0 - m), e1 = __expf(d1 - m), e2 = __expf(d2 - m);
  const float inv = 1.0f / (e0 + e1 + e2);
  out[((long)b * R + r) * H + t] = (e0 * x0 + e1 * x1 + e2 * x2) * inv;
}

struct MHA { const float *WK, *WO, *WQ, *WV, *lb, *bK, *bO, *bQ, *bV, *lg; };
}  // namespace

extern "C" void kernel_launch(void* const* d_in, const int* in_sizes, int n_in,
                              void* d_out, int out_size, void* d_ws,
                              size_t ws_size, hipStream_t stream) {
  (void)in_sizes; (void)n_in; (void)out_size; (void)ws_size;
  const float* char_embeds = (const float*)d_in[0];
  const float* word_embeds = (const float*)d_in[1];
  const float* bmes = (const float*)d_in[2];
  const float* wlen = (const float*)d_in[3];
  const int* n2e_mask = (const int*)d_in[4];
  const int* e2n_mask = (const int*)d_in[5];
  // params pytree, sorted-key flatten order
  int pi = 6;
  auto F = [&](int& i) { return (const float*)d_in[i++]; };
  auto loadMHA = [&](int& i) {
    MHA m;
    m.WK = F(i); m.WO = F(i); m.WQ = F(i); m.WV = F(i);
    m.lb = F(i); m.bK = F(i); m.bO = F(i); m.bQ = F(i); m.bV = F(i); m.lg = F(i);
    return m;
  };
  const float* att_We = F(pi);
  const float* att_Wn = F(pi);
  const float* att_be = F(pi);
  const float* att_bn = F(pi);
  MHA e2n = loadMHA(pi);
  const float* edge_W = F(pi);
  const float* edge_b = F(pi);
  const float* glo_W = F(pi);
  const float* glo_b = F(pi);
  MHA gloe = loadMHA(pi);
  MHA glon = loadMHA(pi);
  const float* ln_b = F(pi);
  const float* ln_g = F(pi);
  MHA n2e = loadMHA(pi);
  const float* node_W = F(pi);
  const float* node_b = F(pi);

  // workspace layout (floats)
  float* ws = (float*)d_ws;
  float* nodes = ws;               ws += RH;
  float* edges = ws;               ws += RH;
  float* glo = ws;                 ws += B * H;
  float* gatt = ws;                ws += B * 2 * H;
  float* q = ws;                   ws += RH;
  float* kbuf = ws;                ws += RH;
  float* vbuf = ws;                ws += RH;
  float* att_out = ws;             ws += RH;
  float* cat = ws;                 ws += (long)RB * 640;
  float* gates = ws;               ws += 4 * RH;
  float* nodes_r = ws;             ws += 3 * RH;
  float* edges_r = ws;             /* ws += 3*RH; */

  const dim3 g1(P / 16, RB / 16, 1);
  constexpr size_t SMEM_N2E =
      (size_t)(2 * L * P + P + NH * L + P + 128 + L * BMES) * sizeof(float);

  ln_rows<<<RB, 128, 0, stream>>>(char_embeds, ln_g, ln_b, nodes);
  ln_rows<<<RB, 128, 0, stream>>>(word_embeds, ln_g, ln_b, edges);
  glo_mean<<<B, 128, 0, stream>>>(nodes, edges, glo);

  for (int i = 0; i < ITERS; ++i) {
    const long sKV_n2e = (long)i * (H + BMES) * P;
    const long sKV_e2n = (long)i * (2 * H + LEN) * P;
    const long sHP = (long)i * H * P;
    // ---- n2e MHA: edges attend over chars ----
    gemm16<false><<<g1, 32, 0, stream>>>(edges, n2e.WQ + sHP, n2e.bQ + i * P, q, H, 0, 0, 0);
    gemm16<false><<<g1, 32, 0, stream>>>(nodes, n2e.WK + sKV_n2e, n2e.bK + i * P, kbuf, H, 0, 0, 0);
    gemm16<false><<<g1, 32, 0, stream>>>(nodes, n2e.WV + sKV_n2e, n2e.bV + i * P, vbuf, H, 0, 0, 0);
    attn_n2e<<<B * E, 256, SMEM_N2E, stream>>>(
        q, kbuf, vbuf, bmes, n2e.WK + sKV_n2e + (long)H * P,
        n2e.WV + sKV_n2e + (long)H * P, n2e_mask, n2e.WO + sHP, n2e.bO + i * H,
        n2e.lg + i * H, n2e.lb + i * H, edges, att_out);
    // ---- global attention over (pre-update) nodes and edges ----
    gemm16<false><<<g1, 32, 0, stream>>>(nodes, glon.WK + sHP, glon.bK + i * P, kbuf, H, 0, 0, 0);
    gemm16<false><<<g1, 32, 0, stream>>>(nodes, glon.WV + sHP, glon.bV + i * P, vbuf, H, 0, 0, 0);
    glo_att<<<B, 256, 0, stream>>>(glo, kbuf, vbuf, glon.WQ + sHP, glon.bQ + i * P,
                                   glon.WO + sHP, glon.bO + i * H, glon.lg + i * H,
                                   glon.lb + i * H, gatt, 0);
    gemm16<false><<<g1, 32, 0, stream>>>(edges, gloe.WK + sHP, gloe.bK + i * P, kbuf, H, 0, 0, 0);
    gemm16<false><<<g1, 32, 0, stream>>>(edges, gloe.WV + sHP, gloe.bV + i * P, vbuf, H, 0, 0, 0);
    glo_att<<<B, 256, 0, stream>>>(glo, kbuf, vbuf, gloe.WQ + sHP, gloe.bQ + i * P,
                                   gloe.WO + sHP, gloe.bO + i * H, gloe.lg + i * H,
                                   gloe.lb + i * H, gatt, H);
    // ---- edge cell ----
    cat_edge<<<(RB * 512 + 255) / 256, 256, 0, stream>>>(edges, att_out, gatt, cat);
    gemm16<false><<<dim3(P / 16, RB / 16, 3), 32, 0, stream>>>(
        cat, edge_W, edge_b, gates, 512, 512L * H, H, RH);
    edge_combine<<<RB, 128, 0, stream>>>(gates, edges, edges_r + (long)i * RH);
    // ---- e2n MHA: chars attend over (updated) edges ----
    gemm16<false><<<g1, 32, 0, stream>>>(nodes, e2n.WQ + sHP, e2n.bQ + i * P, q, H, 0, 0, 0);
    gemm16<false><<<g1, 32, 0, stream>>>(edges, e2n.WK + sKV_e2n, e2n.bK + i * P, kbuf, H, 0, 0, 0);
    gemm16<true><<<g1, 32, 0, stream>>>(word_embeds, e2n.WK + sKV_e2n + (long)H * P,
                                        nullptr, kbuf, H, 0, 0, 0);
    gemm16<false><<<g1, 32, 0, stream>>>(edges, e2n.WV + sKV_e2n, e2n.bV + i * P, vbuf, H, 0, 0, 0);
    gemm16<true><<<g1, 32, 0, stream>>>(word_embeds, e2n.WV + sKV_e2n + (long)H * P,
                                        nullptr, vbuf, H, 0, 0, 0);
    lenproj_add<<<RB, 128, 0, stream>>>(wlen, e2n.WK + sKV_e2n + 2L * H * P,
                                        e2n.WV + sKV_e2n + 2L * H * P, kbuf, vbuf);
    attn_e2n<<<B * L, 256, 0, stream>>>(q, kbuf, vbuf, e2n_mask, e2n.WO + sHP,
                                        e2n.bO + i * H, e2n.lg + i * H,
                                        e2n.lb + i * H, nodes, att_out);
    // ---- node cell ----
    cat_node<<<(RB * 640 + 255) / 256, 256, 0, stream>>>(nodes, char_embeds, att_out, gatt, cat);
    gemm16<false><<<dim3(P / 16, RB / 16, 4), 32, 0, stream>>>(
        cat, node_W, node_b, gates, 640, 640L * H, H, RH);
    node_combine<<<RB, 128, 0, stream>>>(gates, att_out, nodes, nodes_r + (long)i * RH);
    // ---- global cell ----
    glo_cell<<<B, 128, 0, stream>>>(gatt, glo_W, glo_b, glo);
  }

  float* out_nodes = (float*)d_out;
  float* out_edges = out_nodes + RH;
  layer_att<<<B * L, 128, 0, stream>>>(nodes_r, att_Wn, att_bn, out_nodes, L);
  layer_att<<<B * E, 128, 0, stream>>>(edges_r, att_We, att_be, out_edges, E);
}